// GIConv2dWeights_88055419503085
// MI455X (gfx1250) — compile-verified
//
#include <hip/hip_runtime.h>
#include <hip/hip_bf16.h>

// ---------------------------------------------------------------------------
// Problem constants (from reference)
// ---------------------------------------------------------------------------
#define S     4
#define BB    32
#define CIN   128
#define COUT  256
#define HW    256           // H*W = 16*16
#define D     1152          // CIN*3*3
#define NTOT  8192          // BB*HW (gram reduction length)
#define NB    64            // Cholesky block size
#define NBLK  18            // D / NB
#define RW    512           // RHS width: 256 (XLY) + 256 (Z^T)
#define WM_ELEMS ((size_t)S * COUT * D)   // 1,179,648

typedef __attribute__((ext_vector_type(2))) float v2f;
typedef __attribute__((ext_vector_type(8))) float v8f;

__device__ __forceinline__ v8f v8f_zero() {
    v8f z = {0.f, 0.f, 0.f, 0.f, 0.f, 0.f, 0.f, 0.f};
    return z;
}

// D = A(16x4) * B(4x16) + C  with f32 operands.
__device__ __forceinline__ v8f wmma_f32(v2f a, v2f b, v8f c) {
    return __builtin_amdgcn_wmma_f32_16x16x4_f32(
        false, a, false, b, (short)0, c, false, false);
}

// ---------------------------------------------------------------------------
// Patch generator: P[s, d, n] with d = cin*9 + dy*3 + dx, n = b*256 + y*16 + x
// value = sqrt_prec[b] * X[s, b, cin, y+dy-1, x+dx-1]  (zero outside image)
// ---------------------------------------------------------------------------
__device__ __forceinline__ float patch_val(const float* __restrict__ X,
                                           const float* __restrict__ spl,
                                           int s, int d, int n) {
    int cin = d / 9;
    int r   = d - cin * 9;
    int dy  = r / 3;
    int dx  = r - dy * 3;
    int b   = n >> 8;
    int p   = n & 255;
    int y   = (p >> 4) + dy - 1;
    int x   = (p & 15) + dx - 1;
    if ((unsigned)y >= 16u || (unsigned)x >= 16u) return 0.f;
    size_t idx = (((size_t)((s * BB + b) * CIN + cin)) << 8) + (y << 4) + x;
    return spl[b] * X[idx];
}

// ---------------------------------------------------------------------------
// 128x128 WMMA micro-kernel over LDS tiles At[m][k], Bt[n][k] (stride 33),
// 8 waves arranged 4x2; each wave owns a 32x64 subtile (2x4 accumulators).
// Consumes a KC=32 chunk.
// ---------------------------------------------------------------------------
__device__ __forceinline__ void wmma_chunk_128(const float* __restrict__ At,
                                               const float* __restrict__ Bt,
                                               v8f acc[2][4], int lane,
                                               int wm, int wn) {
    const int hi  = lane >> 4;
    const int l15 = lane & 15;
#pragma unroll
    for (int kk = 0; kk < 32; kk += 4) {
        const int ka = kk + (hi << 1);
        v2f a[2], b[4];
#pragma unroll
        for (int mi = 0; mi < 2; ++mi) {
            int m = wm * 32 + mi * 16 + l15;
            a[mi].x = At[m * 33 + ka];
            a[mi].y = At[m * 33 + ka + 1];
        }
#pragma unroll
        for (int ni = 0; ni < 4; ++ni) {
            int n = wn * 64 + ni * 16 + l15;
            b[ni].x = Bt[n * 33 + ka];
            b[ni].y = Bt[n * 33 + ka + 1];
        }
#pragma unroll
        for (int mi = 0; mi < 2; ++mi)
#pragma unroll
            for (int ni = 0; ni < 4; ++ni)
                acc[mi][ni] = wmma_f32(a[mi], b[ni], acc[mi][ni]);
    }
}

// ---------------------------------------------------------------------------
// K0: sqrt_prec[b] = exp(0.5 * log_prec_scaled[b])
// ---------------------------------------------------------------------------
__global__ void k_sqrtprec(const float* __restrict__ lp, float* __restrict__ sp) {
    int t = threadIdx.x;
    if (t < BB) sp[t] = expf(0.5f * lp[t]);
}

// ---------------------------------------------------------------------------
// K1: XLX[s] = P P^T + I  (lower-triangular tiles only), 128x128 tiles.
// grid (9, 9, S), 256 threads. Tiles with jt > it early-exit.
// ---------------------------------------------------------------------------
__global__ __launch_bounds__(256) void kernel_gram(const float* __restrict__ X,
                                                   const float* __restrict__ sp,
                                                   float* __restrict__ Lbuf) {
    const int s  = blockIdx.z;
    const int it = blockIdx.x, jt = blockIdx.y;
    if (jt > it) return;
    const int i0 = it * 128, j0 = jt * 128;

    __shared__ float At[128 * 33];
    __shared__ float Bt[128 * 33];
    __shared__ float spl[BB];

    const int tid = threadIdx.x;
    if (tid < BB) spl[tid] = sp[tid];
    __syncthreads();

    const int lane = tid & 31, wid = tid >> 5;
    const int wm = wid >> 1, wn = wid & 1;

    v8f acc[2][4];
#pragma unroll
    for (int mi = 0; mi < 2; ++mi)
#pragma unroll
        for (int ni = 0; ni < 4; ++ni) acc[mi][ni] = v8f_zero();

    for (int n0 = 0; n0 < NTOT; n0 += 32) {
        __syncthreads();
        for (int idx = tid; idx < 128 * 32; idx += 256) {
            int k = idx & 31, m = idx >> 5;
            At[m * 33 + k] = patch_val(X, spl, s, i0 + m, n0 + k);
            Bt[m * 33 + k] = patch_val(X, spl, s, j0 + m, n0 + k);
        }
        __syncthreads();
        wmma_chunk_128(At, Bt, acc, lane, wm, wn);
    }

    float* Cb = Lbuf + (size_t)s * D * D;
    const int hi = lane >> 4, l15 = lane & 15;
#pragma unroll
    for (int mi = 0; mi < 2; ++mi)
#pragma unroll
        for (int ni = 0; ni < 4; ++ni)
#pragma unroll
            for (int v = 0; v < 8; ++v) {
                int row = i0 + wm * 32 + mi * 16 + v + 8 * hi;
                int col = j0 + wn * 64 + ni * 16 + l15;
                float val = acc[mi][ni][v];
                if (row == col) val += 1.0f;   // PRIOR_PREC
                Cb[(size_t)row * D + col] = val;
            }
}

// ---------------------------------------------------------------------------
// K2: XLY[s] = P * Yf^T  -> written into R[:, :, 0:256].
// grid (9, 2, S), 256 threads.  Yf[b,c,p] = sp[b]*u[b,c,p].
// ---------------------------------------------------------------------------
__global__ __launch_bounds__(256) void kernel_xly(const float* __restrict__ X,
                                                  const float* __restrict__ u,
                                                  const float* __restrict__ sp,
                                                  float* __restrict__ Rbuf) {
    const int s  = blockIdx.z;
    const int it = blockIdx.x, jt = blockIdx.y;
    const int i0 = it * 128, j0 = jt * 128;

    __shared__ float At[128 * 33];
    __shared__ float Bt[128 * 33];
    __shared__ float spl[BB];

    const int tid = threadIdx.x;
    if (tid < BB) spl[tid] = sp[tid];
    __syncthreads();

    const int lane = tid & 31, wid = tid >> 5;
    const int wm = wid >> 1, wn = wid & 1;

    v8f acc[2][4];
#pragma unroll
    for (int mi = 0; mi < 2; ++mi)
#pragma unroll
        for (int ni = 0; ni < 4; ++ni) acc[mi][ni] = v8f_zero();

    for (int n0 = 0; n0 < NTOT; n0 += 32) {
        __syncthreads();
        for (int idx = tid; idx < 128 * 32; idx += 256) {
            int k = idx & 31, m = idx >> 5;
            At[m * 33 + k] = patch_val(X, spl, s, i0 + m, n0 + k);
            int nn = n0 + k;
            int b = nn >> 8, p = nn & 255;
            int c = j0 + m;
            Bt[m * 33 + k] = spl[b] * u[(size_t)(b * COUT + c) * HW + p];
        }
        __syncthreads();
        wmma_chunk_128(At, Bt, acc, lane, wm, wn);
    }

    float* Cb = Rbuf + (size_t)s * D * RW;
    const int hi = lane >> 4, l15 = lane & 15;
#pragma unroll
    for (int mi = 0; mi < 2; ++mi)
#pragma unroll
        for (int ni = 0; ni < 4; ++ni)
#pragma unroll
            for (int v = 0; v < 8; ++v) {
                int row = i0 + wm * 32 + mi * 16 + v + 8 * hi;
                int col = j0 + wn * 64 + ni * 16 + l15;
                Cb[(size_t)row * RW + col] = acc[mi][ni][v];
            }
}

// ---------------------------------------------------------------------------
// K3: R[s, d, 256+c] = Z[s, c, d]
// ---------------------------------------------------------------------------
__global__ void k_ztrans(const float* __restrict__ Z, float* __restrict__ Rbuf) {
    size_t g = (size_t)blockIdx.x * 256 + threadIdx.x;
    if (g >= WM_ELEMS) return;
    int d  = (int)(g % D);
    int sc = (int)(g / D);
    int c  = sc % COUT;
    int s  = sc / COUT;
    Rbuf[(size_t)s * D * RW + (size_t)d * RW + 256 + c] = Z[g];
}

// ---------------------------------------------------------------------------
// Generic 64x64 WMMA GEMM-accumulate:  C_tile -= A(MxK) * B(KxN)
//   transA=1: element A_op[m][k] = A[k*lda + aRow + m]   (A stored K x M)
//   transB=1: element B_op[k][n] = B[(bCol+n)*ldb + k]   (B stored N x K)
// 128 threads (4 waves), each wave owns a 16x64 strip (4 accumulators).
// ---------------------------------------------------------------------------
__global__ __launch_bounds__(128) void gemm_acc(
    float* __restrict__ Cp, long long cS, int ldc,
    const float* __restrict__ Ap, long long aS, int lda, int transA, int aK0,
    const float* __restrict__ Bp, long long bS, int ldb, int transB, int bK0,
    int cR0, int cC0, int Ksz, int triSkip) {
    const int s  = blockIdx.z;
    const int it = cR0 + blockIdx.x, jt = cC0 + blockIdx.y;
    if (triSkip && jt > it) return;

    const float* A = Ap + (size_t)s * aS;
    const float* Bm = Bp + (size_t)s * bS;
    float* Cb = Cp + (size_t)s * cS + (size_t)(it * 64) * ldc + jt * 64;
    const int aR = it * 64, bC = jt * 64;

    __shared__ float At[64 * 33];
    __shared__ float Bt[64 * 33];

    const int tid = threadIdx.x;
    const int lane = tid & 31, wid = tid >> 5;

    v8f acc[4];
#pragma unroll
    for (int ni = 0; ni < 4; ++ni) acc[ni] = v8f_zero();

    for (int k0 = 0; k0 < Ksz; k0 += 32) {
        __syncthreads();
        for (int idx = tid; idx < 64 * 32; idx += 128) {
            int k = idx & 31, m = idx >> 5;
            int kga = aK0 + k0 + k;
            int kgb = bK0 + k0 + k;
            At[m * 33 + k] = transA ? A[(size_t)kga * lda + aR + m]
                                    : A[(size_t)(aR + m) * lda + kga];
            Bt[m * 33 + k] = transB ? Bm[(size_t)(bC + m) * ldb + kgb]
                                    : Bm[(size_t)kgb * ldb + bC + m];
        }
        __syncthreads();

        const int hi = lane >> 4, l15 = lane & 15;
#pragma unroll
        for (int kk = 0; kk < 32; kk += 4) {
            const int ka = kk + (hi << 1);
            v2f a;
            a.x = At[(wid * 16 + l15) * 33 + ka];
            a.y = At[(wid * 16 + l15) * 33 + ka + 1];
            v2f b[4];
#pragma unroll
            for (int ni = 0; ni < 4; ++ni) {
                b[ni].x = Bt[(ni * 16 + l15) * 33 + ka];
                b[ni].y = Bt[(ni * 16 + l15) * 33 + ka + 1];
            }
#pragma unroll
            for (int ni = 0; ni < 4; ++ni) acc[ni] = wmma_f32(a, b[ni], acc[ni]);
        }
    }

    const int hi = lane >> 4, l15 = lane & 15;
#pragma unroll
    for (int ni = 0; ni < 4; ++ni)
#pragma unroll
        for (int v = 0; v < 8; ++v) {
            int row = wid * 16 + v + 8 * hi;
            int col = ni * 16 + l15;
            Cb[(size_t)row * ldc + col] -= acc[ni][v];
        }
}

// ---------------------------------------------------------------------------
// Cholesky: factor 64x64 diagonal block in LDS (lower).  grid (S), 64 thr.
// ---------------------------------------------------------------------------
__global__ __launch_bounds__(64) void chol_diag(float* __restrict__ Lbuf, int kb) {
    const int s = blockIdx.x;
    float* Ag = Lbuf + (size_t)s * D * D + (size_t)(kb * 64) * D + kb * 64;
    __shared__ float T[64 * 65];
    const int tid = threadIdx.x;

    for (int idx = tid; idx < 64 * 64; idx += 64) {
        int r = idx >> 6, c = idx & 63;
        T[r * 65 + c] = Ag[(size_t)r * D + c];
    }
    __syncthreads();

    for (int j = 0; j < 64; ++j) {
        if (tid == 0) T[j * 65 + j] = sqrtf(T[j * 65 + j]);
        __syncthreads();
        float dinv = 1.0f / T[j * 65 + j];
        if (tid > j) T[tid * 65 + j] *= dinv;
        __syncthreads();
        if (tid > j) {
            float lij = T[tid * 65 + j];
            for (int k = j + 1; k <= tid; ++k) T[tid * 65 + k] -= lij * T[k * 65 + j];
        }
        __syncthreads();
    }

    for (int idx = tid; idx < 64 * 64; idx += 64) {
        int r = idx >> 6, c = idx & 63;
        if (c <= r) Ag[(size_t)r * D + c] = T[r * 65 + c];
    }
}

// ---------------------------------------------------------------------------
// Cholesky panel solve: L21 = A21 * L11^{-T}, one 64-row slab per block.
// grid (slabsBelow, S), 64 threads; thread t owns row t of the slab.
// ---------------------------------------------------------------------------
__global__ __launch_bounds__(64) void chol_trsm(float* __restrict__ Lbuf, int kb) {
    const int s  = blockIdx.y;
    const int ib = kb + 1 + blockIdx.x;
    float* base = Lbuf + (size_t)s * D * D;
    const float* Lk = base + (size_t)(kb * 64) * D + kb * 64;
    float* Pg = base + (size_t)(ib * 64) * D + kb * 64;

    __shared__ float Lkk[64 * 65];
    __shared__ float Rw[64 * 65];
    const int tid = threadIdx.x;

    for (int idx = tid; idx < 64 * 64; idx += 64) {
        int r = idx >> 6, c = idx & 63;
        Lkk[r * 65 + c] = Lk[(size_t)r * D + c];
        Rw[r * 65 + c]  = Pg[(size_t)r * D + c];
    }
    __syncthreads();

    // forward substitution on each owned row (independent per thread)
    for (int j = 0; j < 64; ++j) {
        float x = Rw[tid * 65 + j] / Lkk[j * 65 + j];
        Rw[tid * 65 + j] = x;
        for (int t = j + 1; t < 64; ++t) Rw[tid * 65 + t] -= x * Lkk[t * 65 + j];
    }
    __syncthreads();

    for (int idx = tid; idx < 64 * 64; idx += 64) {
        int r = idx >> 6, c = idx & 63;
        Pg[(size_t)r * D + c] = Rw[r * 65 + c];
    }
}

// ---------------------------------------------------------------------------
// Block forward solve: L_kk X = R_k (128 columns per block).
// grid (colTiles, S), 128 threads; thread owns one column.
// ---------------------------------------------------------------------------
__global__ __launch_bounds__(128) void fwd_solve(const float* __restrict__ Lbuf,
                                                 float* __restrict__ Rbuf, int kb) {
    const int s  = blockIdx.y;
    const int c0 = blockIdx.x * 128;
    const float* Lk = Lbuf + (size_t)s * D * D + (size_t)(kb * 64) * D + kb * 64;
    float* Rg = Rbuf + (size_t)s * D * RW + (size_t)(kb * 64) * RW + c0;

    __shared__ float Lkk[64 * 65];
    __shared__ float Xb[64 * 129];
    const int tid = threadIdx.x;

    for (int idx = tid; idx < 64 * 64; idx += 128) {
        int r = idx >> 6, c = idx & 63;
        Lkk[r * 65 + c] = Lk[(size_t)r * D + c];
    }
    for (int j = 0; j < 64; ++j) Xb[j * 129 + tid] = Rg[(size_t)j * RW + tid];
    __syncthreads();

    for (int j = 0; j < 64; ++j) {
        float x = Xb[j * 129 + tid] / Lkk[j * 65 + j];
        Xb[j * 129 + tid] = x;
        for (int t = j + 1; t < 64; ++t) Xb[t * 129 + tid] -= x * Lkk[t * 65 + j];
    }

    for (int j = 0; j < 64; ++j) Rg[(size_t)j * RW + tid] = Xb[j * 129 + tid];
}

// ---------------------------------------------------------------------------
// Block backward solve: L_kk^T X = R_k (upper-triangular substitution).
// ---------------------------------------------------------------------------
__global__ __launch_bounds__(128) void bwd_solve(const float* __restrict__ Lbuf,
                                                 float* __restrict__ Rbuf, int kb) {
    const int s  = blockIdx.y;
    const int c0 = blockIdx.x * 128;
    const float* Lk = Lbuf + (size_t)s * D * D + (size_t)(kb * 64) * D + kb * 64;
    float* Rg = Rbuf + (size_t)s * D * RW + (size_t)(kb * 64) * RW + c0;

    __shared__ float Lkk[64 * 65];
    __shared__ float Xb[64 * 129];
    const int tid = threadIdx.x;

    for (int idx = tid; idx < 64 * 64; idx += 128) {
        int r = idx >> 6, c = idx & 63;
        Lkk[r * 65 + c] = Lk[(size_t)r * D + c];
    }
    for (int j = 0; j < 64; ++j) Xb[j * 129 + tid] = Rg[(size_t)j * RW + tid];
    __syncthreads();

    for (int j = 63; j >= 0; --j) {
        float x = Xb[j * 129 + tid] / Lkk[j * 65 + j];
        Xb[j * 129 + tid] = x;
        for (int t = 0; t < j; ++t) Xb[t * 129 + tid] -= x * Lkk[j * 65 + t];
    }

    for (int j = 0; j < 64; ++j) Rg[(size_t)j * RW + tid] = Xb[j * 129 + tid];
}

// ---------------------------------------------------------------------------
// Finalize part 1: Wm[s,c,d] = R[s,d,c] + R[s,d,256+c]; per-block partial
// sums of Wm^2 and Z^2 (deterministic two-stage reduction, no atomics).
// grid (288, S), 256 threads; 1024 elements per block.
// ---------------------------------------------------------------------------
__global__ __launch_bounds__(256) void k_wm(const float* __restrict__ Rbuf,
                                            const float* __restrict__ Z,
                                            float* __restrict__ out,
                                            float* __restrict__ pWW,
                                            float* __restrict__ pZZ) {
    const int s = blockIdx.y;
    const int base = blockIdx.x * 1024;
    const int tid = threadIdx.x;
    const float* Rs = Rbuf + (size_t)s * D * RW;

    float ww = 0.f, zz = 0.f;
#pragma unroll
    for (int t = 0; t < 4; ++t) {
        int g = base + t * 256 + tid;          // < COUT*D = 294912
        int c = g / D, d = g - c * D;
        float w = Rs[(size_t)d * RW + c] + Rs[(size_t)d * RW + 256 + c];
        out[(size_t)s * (COUT * D) + g] = w;
        float z = Z[(size_t)s * (COUT * D) + g];
        ww += w * w;
        zz += z * z;
    }

    __shared__ float sw[256];
    __shared__ float sz[256];
    sw[tid] = ww; sz[tid] = zz;
    __syncthreads();
    for (int o = 128; o > 0; o >>= 1) {
        if (tid < o) { sw[tid] += sw[tid + o]; sz[tid] += sz[tid + o]; }
        __syncthreads();
    }
    if (tid == 0) {
        pWW[s * 288 + blockIdx.x] = sw[0];
        pZZ[s * 288 + blockIdx.x] = sz[0];
    }
}

// ---------------------------------------------------------------------------
// Finalize part 2: logdet from diag(L), then logpq[s].
// logpq = -0.5*sum(Wm^2) + 0.5*sum(Z^2) - 0.5*COUT*logdet,
// logdet = 2*sum(log diag(L))  ->  logpq = -0.5 ww + 0.5 zz - 256 * sumlog
// ---------------------------------------------------------------------------
__global__ __launch_bounds__(256) void k_final(const float* __restrict__ Lbuf,
                                               const float* __restrict__ pWW,
                                               const float* __restrict__ pZZ,
                                               float* __restrict__ out) {
    const int s = blockIdx.x, tid = threadIdx.x;
    const float* Lb = Lbuf + (size_t)s * D * D;
    float t = 0.f;
    for (int d = tid; d < D; d += 256) t += logf(Lb[(size_t)d * D + d]);
    __shared__ float red[256];
    red[tid] = t;
    __syncthreads();
    for (int o = 128; o > 0; o >>= 1) {
        if (tid < o) red[tid] += red[tid + o];
        __syncthreads();
    }
    if (tid == 0) {
        float sumlog = red[0];
        float ww = 0.f, zz = 0.f;
        for (int i = 0; i < 288; ++i) {
            ww += pWW[s * 288 + i];
            zz += pZZ[s * 288 + i];
        }
        out[WM_ELEMS + s] = -0.5f * ww + 0.5f * zz - 256.0f * sumlog;
    }
}

// ---------------------------------------------------------------------------
// Host orchestration
// ---------------------------------------------------------------------------
extern "C" void kernel_launch(void* const* d_in, const int* in_sizes, int n_in,
                              void* d_out, int out_size, void* d_ws, size_t ws_size,
                              hipStream_t stream) {
    const float* X  = (const float*)d_in[0];   // (S,B,CIN,H,W)
    const float* u  = (const float*)d_in[1];   // (B,COUT,H,W)
    const float* lp = (const float*)d_in[2];   // (B,)
    const float* Z  = (const float*)d_in[3];   // (S,COUT,D)
    float* out = (float*)d_out;

    float* ws  = (float*)d_ws;
    float* Lb  = ws;                                   // S*D*D
    float* Rb  = Lb + (size_t)S * D * D;               // S*D*RW
    float* sp  = Rb + (size_t)S * D * RW;              // 64
    float* pWW = sp + 64;                              // S*288
    float* pZZ = pWW + (size_t)S * 288;                // S*288

    // 1) scale factors
    k_sqrtprec<<<1, 32, 0, stream>>>(lp, sp);

    // 2) XLX + I (lower tiles), XLY, Z^T staging
    kernel_gram<<<dim3(9, 9, S), 256, 0, stream>>>(X, sp, Lb);
    kernel_xly<<<dim3(9, 2, S), 256, 0, stream>>>(X, u, sp, Rb);
    k_ztrans<<<(unsigned)((WM_ELEMS + 255) / 256), 256, 0, stream>>>(Z, Rb);

    // 3) blocked Cholesky (right-looking), NB=64
    for (int kb = 0; kb < NBLK; ++kb) {
        chol_diag<<<S, 64, 0, stream>>>(Lb, kb);
        int m = NBLK - 1 - kb;
        if (m > 0) {
            chol_trsm<<<dim3(m, S), 64, 0, stream>>>(Lb, kb);
            // SYRK: A22 -= L21 * L21^T (lower tiles only)
            gemm_acc<<<dim3(m, m, S), 128, 0, stream>>>(
                Lb, (long long)D * D, D,
                Lb, (long long)D * D, D, 0, kb * 64,
                Lb, (long long)D * D, D, 1, kb * 64,
                kb + 1, kb + 1, 64, 1);
        }
    }

    // 4) forward solve L Y = XLY (columns 0..255 of R)
    for (int kb = 0; kb < NBLK; ++kb) {
        fwd_solve<<<dim3(2, S), 128, 0, stream>>>(Lb, Rb, kb);
        int m = NBLK - 1 - kb;
        if (m > 0) {
            gemm_acc<<<dim3(m, 4, S), 128, 0, stream>>>(
                Rb, (long long)D * RW, RW,
                Lb, (long long)D * D, D, 0, kb * 64,
                Rb, (long long)D * RW, RW, 0, kb * 64,
                kb + 1, 0, 64, 0);
        }
    }

    // 5) backward solve L^T [mu | dW'] = [Y | Z^T] (all 512 columns)
    for (int kb = NBLK - 1; kb >= 0; --kb) {
        bwd_solve<<<dim3(4, S), 128, 0, stream>>>(Lb, Rb, kb);
        if (kb > 0) {
            gemm_acc<<<dim3(kb, 8, S), 128, 0, stream>>>(
                Rb, (long long)D * RW, RW,
                Lb, (long long)D * D, D, 1, kb * 64,
                Rb, (long long)D * RW, RW, 0, kb * 64,
                0, 0, 64, 0);
        }
    }

    // 6) Wm + deterministic reductions + logpq
    k_wm<<<dim3(288, S), 256, 0, stream>>>(Rb, Z, out, pWW, pZZ);
    k_final<<<S, 256, 0, stream>>>(Lb, pWW, pZZ, out);
}